// GCNBaseline_76312978915556
// MI455X (gfx1250) — compile-verified
//
#include <hip/hip_runtime.h>

typedef __attribute__((ext_vector_type(2))) float v2f;
typedef __attribute__((ext_vector_type(8))) float v8f;

#if defined(__gfx1250__) && __has_builtin(__builtin_amdgcn_wmma_f32_16x16x4_f32)
#define GCN_USE_WMMA 1
#else
#define GCN_USE_WMMA 0
#endif

#define WAVES_PER_BLOCK 4

// ---------------- degree / dinv ----------------
__global__ void gcn_deg_init(float* __restrict__ deg, int n) {
    int i = blockIdx.x * blockDim.x + threadIdx.x;
    if (i < n) deg[i] = 1.0f;   // self-loop
}

__global__ void gcn_deg_count(const int* __restrict__ dst, float* __restrict__ deg, int E) {
    int e = blockIdx.x * blockDim.x + threadIdx.x;
    if (e < E) atomicAdd(&deg[dst[e]], 1.0f);
}

__global__ void gcn_make_dinv(float* __restrict__ deg, int n) {
    int i = blockIdx.x * blockDim.x + threadIdx.x;
    if (i < n) deg[i] = rsqrtf(deg[i]);   // deg >= 1 always
}

// ---------------- WMMA GEMM: out[r, 0:64] = dinv[r] * (X[r,:] @ W) ----------------
// X: [n, K] row-major, W: [K, 64] row-major, out: [n, 64].
// One wave -> 16 rows x 64 cols (4 accumulators of 16x16), K stepped by 4.
template <int K>
__global__ __launch_bounds__(WAVES_PER_BLOCK * 32)
void gcn_gemm_wmma(const float* __restrict__ X, const float* __restrict__ W,
                   const float* __restrict__ dinv, float* __restrict__ out, int n) {
    __shared__ float sW[K * 64];
    for (int i = threadIdx.x; i < K * 64; i += blockDim.x) sW[i] = W[i];
    __syncthreads();

    const int wave = threadIdx.x >> 5;
    const int lane = threadIdx.x & 31;
    const int mBase = (blockIdx.x * WAVES_PER_BLOCK + wave) * 16;
    if (mBase >= n) return;           // uniform per wave: EXEC stays all-ones

    const int m  = lane & 15;         // row within A-tile / col within B-,C-tile
    const int hf = lane >> 4;         // lane half selects K pair (A) / K pair (B) / +8 rows (C)

    const float* xrow = X + (size_t)(mBase + m) * K + 2 * hf;

    v8f acc[4] = {};
    for (int k0 = 0; k0 < K; k0 += 4) {
#if GCN_USE_WMMA
        // A 16x4 f32: VGPR0 = K = {k0 | k0+2}, VGPR1 = K = {k0+1 | k0+3} by lane half
        v2f a;
        a.x = xrow[k0];
        a.y = xrow[k0 + 1];
#pragma unroll
        for (int nt = 0; nt < 4; ++nt) {
            const int col = nt * 16 + m;
            // B 4x16 f32: VGPR0 = row {k0 | k0+2}, VGPR1 = row {k0+1 | k0+3} by lane half
            v2f b;
            b.x = sW[(k0 + 2 * hf) * 64 + col];
            b.y = sW[(k0 + 2 * hf + 1) * 64 + col];
            acc[nt] = __builtin_amdgcn_wmma_f32_16x16x4_f32(
                false, a, false, b, (short)0, acc[nt], false, false);
        }
#else
        // scalar fallback (host pass / missing builtin) — same C layout
#pragma unroll
        for (int nt = 0; nt < 4; ++nt) {
            const int col = nt * 16 + m;
#pragma unroll
            for (int v = 0; v < 8; ++v) {
                const float* xr = X + (size_t)(mBase + v + 8 * hf) * K;
                float s = acc[nt][v];
                s += xr[k0 + 0] * sW[(k0 + 0) * 64 + col];
                s += xr[k0 + 1] * sW[(k0 + 1) * 64 + col];
                s += xr[k0 + 2] * sW[(k0 + 2) * 64 + col];
                s += xr[k0 + 3] * sW[(k0 + 3) * 64 + col];
                acc[nt][v] = s;
            }
        }
#endif
    }

    // C/D layout: VGPR v -> row (v + 8*hf), col = lane&15 within tile.
#pragma unroll
    for (int nt = 0; nt < 4; ++nt) {
        const int col = nt * 16 + m;
#pragma unroll
        for (int v = 0; v < 8; ++v) {
            const int row = mBase + v + 8 * hf;
            out[(size_t)row * 64 + col] = dinv[row] * acc[nt][v];
        }
    }
}

// ---------------- edge scatter-add: agg[dst] += hs[src] ----------------
// One thread handles one edge x 4 features (float4 gather + 4 f32 atomics).
__global__ void gcn_scatter_add(const int* __restrict__ src, const int* __restrict__ dst,
                                const float* __restrict__ hs, float* __restrict__ agg,
                                int E) {
    long tid = (long)blockIdx.x * blockDim.x + threadIdx.x;
    long total = (long)E * 16;
    if (tid >= total) return;
    int e  = (int)(tid >> 4);
    int f4 = (int)(tid & 15) * 4;
    int s = src[e];
    int d = dst[e];
    float4 v = *(const float4*)(hs + (size_t)s * 64 + f4);
    float* base = agg + (size_t)d * 64 + f4;
    atomicAdd(base + 0, v.x);
    atomicAdd(base + 1, v.y);
    atomicAdd(base + 2, v.z);
    atomicAdd(base + 3, v.w);
}

// ---------------- epilogue: agg = relu(dinv[row]*agg + b[col]) in place ----------------
__global__ void gcn_finish(const float* __restrict__ dinv, const float* __restrict__ b,
                           float* __restrict__ agg, int n) {
    long t = (long)blockIdx.x * blockDim.x + threadIdx.x;
    if (t >= (long)n * 64) return;
    int i = (int)(t >> 6);
    int f = (int)(t & 63);
    float v = dinv[i] * agg[t] + b[f];
    agg[t] = v > 0.0f ? v : 0.0f;
}

// ---------------- final projection: out[i] = h[i,:] . Wc + bc ----------------
__global__ void gcn_final_proj(const float* __restrict__ h, const float* __restrict__ Wc,
                               const float* __restrict__ bc, float* __restrict__ out, int n) {
    int i = blockIdx.x * blockDim.x + threadIdx.x;
    if (i >= n) return;
    const float* row = h + (size_t)i * 64;
    float acc = 0.0f;
#pragma unroll
    for (int f = 0; f < 64; ++f) acc += row[f] * Wc[f];
    out[i] = acc + bc[0];
}

extern "C" void kernel_launch(void* const* d_in, const int* in_sizes, int n_in,
                              void* d_out, int out_size, void* d_ws, size_t ws_size,
                              hipStream_t stream) {
    const float* x  = (const float*)d_in[0];
    const int*   ei = (const int*)d_in[1];
    const float* W1 = (const float*)d_in[2];
    const float* b1 = (const float*)d_in[3];
    const float* W2 = (const float*)d_in[4];
    const float* b2 = (const float*)d_in[5];
    const float* Wc = (const float*)d_in[6];
    const float* bc = (const float*)d_in[7];
    float* out = (float*)d_out;

    const int N = in_sizes[0] / 128;
    const int E = in_sizes[1] / 2;
    const int* srcIdx = ei;        // edge_index[0]
    const int* dstIdx = ei + E;    // edge_index[1]

    // workspace: dinv [N] | bufA [N*64] | bufB [N*64]  (~51.6 MB)
    float* dinv = (float*)d_ws;
    float* bufA = dinv + N;
    float* bufB = bufA + (size_t)N * 64;

    const int T = 256;
    const int gemmBlocks = (N / 16 + WAVES_PER_BLOCK - 1) / WAVES_PER_BLOCK;
    const long edgeWork = (long)E * 16;
    const long featWork = (long)N * 64;

    // degrees (with self-loops) -> dinv
    gcn_deg_init<<<(N + T - 1) / T, T, 0, stream>>>(dinv, N);
    gcn_deg_count<<<(E + T - 1) / T, T, 0, stream>>>(dstIdx, dinv, E);
    gcn_make_dinv<<<(N + T - 1) / T, T, 0, stream>>>(dinv, N);

    // layer 1: hs1 = dinv * (x @ W1); agg seeded with self-loop term
    gcn_gemm_wmma<128><<<gemmBlocks, WAVES_PER_BLOCK * 32, 0, stream>>>(x, W1, dinv, bufA, N);
    hipMemcpyAsync(bufB, bufA, (size_t)N * 64 * sizeof(float), hipMemcpyDeviceToDevice, stream);
    gcn_scatter_add<<<(int)((edgeWork + T - 1) / T), T, 0, stream>>>(srcIdx, dstIdx, bufA, bufB, E);
    gcn_finish<<<(int)((featWork + T - 1) / T), T, 0, stream>>>(dinv, b1, bufB, N);

    // layer 2: hs2 = dinv * (h1 @ W2)
    gcn_gemm_wmma<64><<<gemmBlocks, WAVES_PER_BLOCK * 32, 0, stream>>>(bufB, W2, dinv, bufA, N);
    hipMemcpyAsync(bufB, bufA, (size_t)N * 64 * sizeof(float), hipMemcpyDeviceToDevice, stream);
    gcn_scatter_add<<<(int)((edgeWork + T - 1) / T), T, 0, stream>>>(srcIdx, dstIdx, bufA, bufB, E);
    gcn_finish<<<(int)((featWork + T - 1) / T), T, 0, stream>>>(dinv, b2, bufB, N);

    // classifier head
    gcn_final_proj<<<(N + T - 1) / T, T, 0, stream>>>(bufB, Wc, bc, out, N);
}